// GraphSAGEEncoder_69810398429749
// MI455X (gfx1250) — compile-verified
//
#include <hip/hip_runtime.h>
#include <hip/hip_bf16.h>

#define NNODES  100000
#define NEDGES  1600000
#define DFEAT   64
#define NGRAPHS 256
#define BN_EPS  1e-5f

typedef __attribute__((ext_vector_type(2))) float v2f;
typedef __attribute__((ext_vector_type(8))) float v8f;

// ---------------------------------------------------------------------------
// zero a float buffer (grid-stride)
__global__ void zero_kernel(float* __restrict__ p, int n) {
    for (int i = blockIdx.x * blockDim.x + threadIdx.x; i < n;
         i += gridDim.x * blockDim.x)
        p[i] = 0.0f;
}

// ---------------------------------------------------------------------------
// deg[dst] += 1 per edge
__global__ void degree_kernel(const long long* __restrict__ dst,
                              float* __restrict__ deg, int nEdges) {
    int e = blockIdx.x * blockDim.x + threadIdx.x;
    if (e < nEdges) atomicAdd(&deg[dst[e]], 1.0f);
}

// ---------------------------------------------------------------------------
// agg[dst] += feat[src]; 16 threads/edge, float4 per thread
__global__ void aggregate_kernel(const float* __restrict__ feat,
                                 const long long* __restrict__ src,
                                 const long long* __restrict__ dst,
                                 float* __restrict__ agg, int nEdges) {
    int t = blockIdx.x * blockDim.x + threadIdx.x;
    int e = t >> 4;
    int f = (t & 15) << 2;
    if (e >= nEdges) return;
    long long s = src[e];
    long long d = dst[e];
    float4 v = *(const float4*)(feat + (size_t)s * DFEAT + f);
    float* p = agg + (size_t)d * DFEAT + f;
    atomicAdd(p + 0, v.x);
    atomicAdd(p + 1, v.y);
    atomicAdd(p + 2, v.z);
    atomicAdd(p + 3, v.w);
}

// ---------------------------------------------------------------------------
// SAGE dense projection via V_WMMA_F32_16X16X4_F32 (exact f32 matmul).
//   out[n,o] = sum_k (agg[n,k]/max(deg,1))*Wl[o,k] + sum_k root[n,k]*Wr[o,k] + bias[o]
//
// One wave computes a full 16-node x 64-output stripe (4 accumulator tiles).
// K-permutation trick: each lane loads float4 at kb = k0 + 4*(lane>>4).
// WMMA K-slots then carry logical k {k0,k0+1,k0+4,k0+5} (.xy issue) and
// {k0+2,k0+3,k0+6,k0+7} (.zw issue) -- identical permutation on A and B, so
// the contraction is exact, and all operand loads become global_load_b128.
// D layout: reg r -> (row = r + (lane>>4)*8, col = oTile*16 + (lane&15)).
// POOL epilogue fuses the global-mean-pool scatter for layer 2.
template <bool POOL>
__global__ __launch_bounds__(256) void sage_gemm_kernel(
    const float* __restrict__ aggbuf, const float* __restrict__ deg,
    const float* __restrict__ root, const float* __restrict__ Wl,
    const float* __restrict__ Wr, const float* __restrict__ bias,
    float* __restrict__ out, const long long* __restrict__ batch,
    float* __restrict__ pooled) {
    const int lane = threadIdx.x;                  // 0..31
    const int tile = blockIdx.x * 8 + threadIdx.y; // node tile (wave-uniform)
    if (tile >= NNODES / 16) return;               // whole wave exits together
    const int nodeBase = tile * 16;
    const int m  = lane & 15;
    const int kh = lane >> 4;

    const int arow = nodeBase + m;
    const float dinv = 1.0f / fmaxf(deg[arow], 1.0f);

    const float* __restrict__ ar  = aggbuf + (size_t)arow * DFEAT;
    const float* __restrict__ xr  = root + (size_t)arow * DFEAT;
    const float* __restrict__ wlb = Wl + (size_t)m * DFEAT; // row for oTile 0
    const float* __restrict__ wrb = Wr + (size_t)m * DFEAT;

    v8f acc[4];
#pragma unroll
    for (int ot = 0; ot < 4; ++ot) acc[ot] = (v8f){};

#pragma unroll
    for (int k0 = 0; k0 < DFEAT; k0 += 8) {
        const int kb = k0 + 4 * kh;
        float4 aA = *(const float4*)(ar + kb);
        float4 aX = *(const float4*)(xr + kb);
        aA.x *= dinv; aA.y *= dinv; aA.z *= dinv; aA.w *= dinv;
        const v2f aA01 = {aA.x, aA.y}, aA23 = {aA.z, aA.w};
        const v2f aX01 = {aX.x, aX.y}, aX23 = {aX.z, aX.w};
#pragma unroll
        for (int ot = 0; ot < 4; ++ot) {
            const size_t wo = (size_t)ot * 16 * DFEAT + kb;
            float4 bL = *(const float4*)(wlb + wo);
            float4 bR = *(const float4*)(wrb + wo);
            const v2f bL01 = {bL.x, bL.y}, bL23 = {bL.z, bL.w};
            const v2f bR01 = {bR.x, bR.y}, bR23 = {bR.z, bR.w};
            acc[ot] = __builtin_amdgcn_wmma_f32_16x16x4_f32(
                false, aA01, false, bL01, (short)0, acc[ot], false, false);
            acc[ot] = __builtin_amdgcn_wmma_f32_16x16x4_f32(
                false, aA23, false, bL23, (short)0, acc[ot], false, false);
            acc[ot] = __builtin_amdgcn_wmma_f32_16x16x4_f32(
                false, aX01, false, bR01, (short)0, acc[ot], false, false);
            acc[ot] = __builtin_amdgcn_wmma_f32_16x16x4_f32(
                false, aX23, false, bR23, (short)0, acc[ot], false, false);
        }
    }

    float bv[4];
#pragma unroll
    for (int ot = 0; ot < 4; ++ot) bv[ot] = bias[ot * 16 + m];

#pragma unroll
    for (int r = 0; r < 8; ++r) {
        const int row = nodeBase + r + kh * 8;
        long long g = 0;
        if (POOL) g = batch[row];
#pragma unroll
        for (int ot = 0; ot < 4; ++ot) {
            const int col = ot * 16 + m;
            const float v = acc[ot][r] + bv[ot];
            if (POOL) {
                atomicAdd(&pooled[(size_t)g * DFEAT + col], v);
            } else {
                out[(size_t)row * DFEAT + col] = v;
            }
        }
    }
}

// ---------------------------------------------------------------------------
// per-column sum / sum-of-squares (64 threads own one column each)
__global__ void bn_stats_kernel(const float* __restrict__ h,
                                float* __restrict__ sum,
                                float* __restrict__ sumsq) {
    const int c = threadIdx.x; // 0..63
    float s = 0.0f, q = 0.0f;
    for (int n = blockIdx.x; n < NNODES; n += gridDim.x) {
        float v = h[(size_t)n * DFEAT + c];
        s += v;
        q += v * v;
    }
    atomicAdd(&sum[c], s);
    atomicAdd(&sumsq[c], q);
}

__global__ void bn_finalize_kernel(const float* __restrict__ sum,
                                   const float* __restrict__ sumsq,
                                   const float* __restrict__ gamma,
                                   const float* __restrict__ beta,
                                   float* __restrict__ scale,
                                   float* __restrict__ shift) {
    const int c = threadIdx.x;
    const float inv_n = 1.0f / (float)NNODES;
    float mu = sum[c] * inv_n;
    float var = sumsq[c] * inv_n - mu * mu;
    float s = gamma[c] * rsqrtf(var + BN_EPS);
    scale[c] = s;
    shift[c] = beta[c] - mu * s;
}

__global__ void bn_relu_kernel(float* __restrict__ h,
                               const float* __restrict__ scale,
                               const float* __restrict__ shift) {
    const int total = NNODES * DFEAT;
    for (int i = blockIdx.x * blockDim.x + threadIdx.x; i < total;
         i += gridDim.x * blockDim.x) {
        int c = i & (DFEAT - 1);
        float v = h[i] * scale[c] + shift[c];
        h[i] = fmaxf(v, 0.0f);
    }
}

// ---------------------------------------------------------------------------
__global__ void cnt_kernel(const long long* __restrict__ batch,
                           float* __restrict__ cnt) {
    int n = blockIdx.x * blockDim.x + threadIdx.x;
    if (n < NNODES) atomicAdd(&cnt[batch[n]], 1.0f);
}

__global__ void pool_div_kernel(const float* __restrict__ pooled,
                                const float* __restrict__ cnt,
                                float* __restrict__ out) {
    int i = blockIdx.x * blockDim.x + threadIdx.x;
    if (i < NGRAPHS * DFEAT)
        out[i] = pooled[i] / fmaxf(cnt[i >> 6], 1.0f);
}

// ---------------------------------------------------------------------------
static inline size_t align256(size_t x) { return (x + 255) & ~(size_t)255; }

extern "C" void kernel_launch(void* const* d_in, const int* in_sizes, int n_in,
                              void* d_out, int out_size, void* d_ws,
                              size_t ws_size, hipStream_t stream) {
    const float* x        = (const float*)d_in[0];
    const long long* eidx = (const long long*)d_in[1];
    const long long* batch= (const long long*)d_in[2];
    const float* W1l = (const float*)d_in[3];
    const float* W1r = (const float*)d_in[4];
    const float* b1  = (const float*)d_in[5];
    const float* gamma = (const float*)d_in[6];
    const float* beta  = (const float*)d_in[7];
    const float* W2l = (const float*)d_in[8];
    const float* W2r = (const float*)d_in[9];
    const float* b2  = (const float*)d_in[10];
    float* out = (float*)d_out;

    const int nEdges = in_sizes[1] / 2; // 1,600,000
    const long long* src = eidx;
    const long long* dst = eidx + nEdges;

    // workspace carve-up
    char* ws = (char*)d_ws;
    size_t off = 0;
    float* deg = (float*)(ws + off); off += align256((size_t)NNODES * 4);
    float* agg = (float*)(ws + off); off += align256((size_t)NNODES * DFEAT * 4);
    float* h   = (float*)(ws + off); off += align256((size_t)NNODES * DFEAT * 4);
    float* stats = (float*)(ws + off); // contiguous accumulator block
    float* bnsum   = stats;            // 64
    float* bnsumsq = stats + 64;       // 64
    float* scale   = stats + 128;      // 64
    float* shift   = stats + 192;      // 64
    float* cnt     = stats + 256;      // 256
    float* pooled  = stats + 512;      // 256*64
    const int statsFloats = 512 + NGRAPHS * DFEAT;

    const int NF = NNODES * DFEAT;
    dim3 blk256(256);

    // ---- zero accumulators (every call: graph-replay safe) ----
    zero_kernel<<<1024, blk256, 0, stream>>>(deg, NNODES);
    zero_kernel<<<4096, blk256, 0, stream>>>(agg, NF);
    zero_kernel<<<8, blk256, 0, stream>>>(stats, statsFloats);

    // ---- degrees (same for both layers) ----
    degree_kernel<<<(nEdges + 255) / 256, blk256, 0, stream>>>(dst, deg, nEdges);

    // ---- layer 1: aggregate -> GEMM(WMMA f32) -> BN stats/apply + ReLU ----
    {
        int tot = nEdges * 16;
        aggregate_kernel<<<(tot + 255) / 256, blk256, 0, stream>>>(x, src, dst, agg, nEdges);
    }
    {
        dim3 blk(32, 8);
        dim3 grd((NNODES / 16 + 7) / 8);
        sage_gemm_kernel<false><<<grd, blk, 0, stream>>>(
            agg, deg, x, W1l, W1r, b1, h, nullptr, nullptr);
    }
    bn_stats_kernel<<<512, dim3(64), 0, stream>>>(h, bnsum, bnsumsq);
    bn_finalize_kernel<<<1, dim3(64), 0, stream>>>(bnsum, bnsumsq, gamma, beta, scale, shift);
    bn_relu_kernel<<<4096, blk256, 0, stream>>>(h, scale, shift);

    // ---- layer 2: re-aggregate -> GEMM(WMMA f32) with fused mean-pool ----
    zero_kernel<<<4096, blk256, 0, stream>>>(agg, NF);
    {
        int tot = nEdges * 16;
        aggregate_kernel<<<(tot + 255) / 256, blk256, 0, stream>>>(h, src, dst, agg, nEdges);
    }
    cnt_kernel<<<(NNODES + 255) / 256, blk256, 0, stream>>>(batch, cnt);
    {
        dim3 blk(32, 8);
        dim3 grd((NNODES / 16 + 7) / 8);
        sage_gemm_kernel<true><<<grd, blk, 0, stream>>>(
            agg, deg, h, W2l, W2r, b2, nullptr, batch, pooled);
    }
    pool_div_kernel<<<(NGRAPHS * DFEAT + 255) / 256, blk256, 0, stream>>>(pooled, cnt, out);
}